// FourierHead2D_35639638622760
// MI455X (gfx1250) — compile-verified
//
#include <hip/hip_runtime.h>
#include <hip/hip_bf16.h>
#include <math.h>

// FourierHead2D, fused:  pdf[b] = |S(t_b)|^2 / (4 * centre_b)
//   S  = sum_{r,c} (pr + i*pi) * e^{i*pi*(c*tx + r*ty)}
//   centre = sum (pr^2 + pi^2)
// where p = X @ W^T + bias, pr = p[:289] as 17x17, pi = p[289:578].

typedef __attribute__((ext_vector_type(2))) float v2f;
typedef __attribute__((ext_vector_type(8))) float v8f;

#define DIMK  512
#define BATCH 32768
#define HSZ   17
#define NPAR  289           // 17*17
#define NOUT  578           // 2*NPAR
#define NT    37            // ceil(578/16) N-tiles of 16
#define LDA   516           // padded LDS row stride in floats (516 % 64 == 4 -> conflict free)

__global__ __launch_bounds__(256, 2)
void fourier_head2d_fused(const float* __restrict__ X,     // (B, 512)
                          const float* __restrict__ T,     // (B, 2)
                          const float* __restrict__ W,     // (578, 512)
                          const float* __restrict__ bias,  // (578,)
                          float* __restrict__ out)         // (B,)
{
    __shared__ float sA[16 * LDA];   // 16 x 512 tile of X, padded rows
    __shared__ float sTgt[32];       // 16 samples x (tx, ty)
    __shared__ float sAcc[48];       // per sample: Sre, Sim, centre

    const int tid     = threadIdx.x;
    const int lane    = tid & 31;
    const int wave    = tid >> 5;
    const int rowBase = blockIdx.x * 16;       // first sample of this block

    if (tid < 48) sAcc[tid] = 0.0f;
    if (tid < 32) sTgt[tid] = T[rowBase * 2 + tid];

    // Cooperatively stage the 16x512 X tile (float4 loads, coalesced).
    for (int i = tid; i < 16 * (DIMK / 4); i += 256) {
        const int r  = i >> 7;       // / 128 float4s per row
        const int c4 = i & 127;
        const float4 v = reinterpret_cast<const float4*>(
                             X + (size_t)(rowBase + r) * DIMK)[c4];
        float* dst = &sA[r * LDA + c4 * 4];
        dst[0] = v.x; dst[1] = v.y; dst[2] = v.z; dst[3] = v.w;
    }
    __syncthreads();

    const int half = lane >> 4;      // 0: lanes 0-15, 1: lanes 16-31
    const int sub  = lane & 15;

    // Each wave owns N-tiles  wave, wave+8, ... (uniform trip count per wave,
    // so EXEC is all-ones at every WMMA as required).
    for (int nt = wave; nt < NT; nt += 8) {
        const int n  = nt * 16 + sub;                 // output column for this lane
        const int nc = (n < NOUT) ? n : (NOUT - 1);   // clamp to stay in-bounds
        const float* wrow = W  + (size_t)nc * DIMK + half * 2;
        const float* arow = sA + sub * LDA          + half * 2;

        v8f acc = {};
        // K = 512 in steps of 4: 128 x v_wmma_f32_16x16x4_f32
#pragma unroll 4
        for (int k0 = 0; k0 < DIMK; k0 += 4) {
            // A frag: lane L, vgpr v -> A[M=L%16][K = k0 + 2*(L/16) + v]
            v2f a = *reinterpret_cast<const v2f*>(arow + k0);
            // B frag: lane L, vgpr v -> B[K = k0 + 2*(L/16) + v][N=L%16] = W[n][k]
            v2f b = *reinterpret_cast<const v2f*>(wrow + k0);
            acc = __builtin_amdgcn_wmma_f32_16x16x4_f32(
                      false, a, false, b, (short)0, acc, false, false);
        }

        // ---- fused epilogue: fold this 16x16 tile of p into (Sre, Sim, centre) ----
        const bool  valid  = (n < NOUT);
        const float bn     = valid ? bias[n] : 0.0f;
        const int   idx    = valid ? ((n < NPAR) ? n : n - NPAR) : 0;
        const bool  isImag = valid && (n >= NPAR);
        const int   fr     = idx / HSZ;          // frequency row  (multiplies ty)
        const int   fc     = idx - fr * HSZ;     // frequency col  (multiplies tx)
        const float frf    = (float)fr;
        const float fcf    = (float)fc;

#pragma unroll
        for (int j = 0; j < 8; ++j) {
            const int   m  = j + 8 * half;       // sample row within tile
            const float v  = acc[j] + bn;        // p[rowBase+m][n]
            const float tx = sTgt[2 * m + 0];
            const float ty = sTgt[2 * m + 1];
            const float u  = fcf * tx + frf * ty;   // phase / pi
            float su, cu;
            sincospif(u, &su, &cu);

            float sre, sim, cen;
            if (isImag) { sre = -v * su; sim =  v * cu; }
            else        { sre =  v * cu; sim =  v * su; }
            cen = v * v;
            if (!valid) { sre = 0.0f; sim = 0.0f; cen = 0.0f; }

            // Reduce across the 16 lanes that share sample row m.
#pragma unroll
            for (int d = 1; d < 16; d <<= 1) {
                sre += __shfl_xor(sre, d, 16);
                sim += __shfl_xor(sim, d, 16);
                cen += __shfl_xor(cen, d, 16);
            }
            if (sub == 0) {                       // lanes 0 and 16 (rows j, j+8)
                atomicAdd(&sAcc[3 * m + 0], sre); // ds_add_f32
                atomicAdd(&sAcc[3 * m + 1], sim);
                atomicAdd(&sAcc[3 * m + 2], cen);
            }
        }
    }
    __syncthreads();

    if (tid < 16) {
        const float sre = sAcc[3 * tid + 0];
        const float sim = sAcc[3 * tid + 1];
        const float cen = sAcc[3 * tid + 2];
        out[rowBase + tid] = (sre * sre + sim * sim) / (4.0f * cen);
    }
}

extern "C" void kernel_launch(void* const* d_in, const int* in_sizes, int n_in,
                              void* d_out, int out_size, void* d_ws, size_t ws_size,
                              hipStream_t stream) {
    (void)in_sizes; (void)n_in; (void)d_ws; (void)ws_size; (void)out_size;
    const float* X  = (const float*)d_in[0];   // inputs  (32768, 512)
    const float* T  = (const float*)d_in[1];   // targets (32768, 2)
    const float* W  = (const float*)d_in[2];   // weight  (578, 512)
    const float* b  = (const float*)d_in[3];   // bias    (578,)
    float* out = (float*)d_out;                // (32768,)

    dim3 grid(BATCH / 16), block(256);
    hipLaunchKernelGGL(fourier_head2d_fused, grid, block, 0, stream, X, T, W, b, out);
}